// Conv2D_79568564125923
// MI455X (gfx1250) — compile-verified
//
#include <hip/hip_runtime.h>

// Problem constants (fixed by the reference)
#define IN_W   4096
#define IN_H   4096
#define OUT_W  4094
#define OUT_H  4094

// Tiling
#define TILE      64            // output tile is TILE x TILE
#define IN_ROWS   66            // TILE + 2 halo rows
#define CHUNKS    17            // float4 chunks per staged row (68 floats >= 66 needed)
#define LSTR      68            // LDS row stride in floats (17 * 4)
#define NTHREADS  256

typedef int          v4i __attribute__((vector_size(16)));
typedef unsigned int v4u __attribute__((vector_size(16)));
typedef int          v8i __attribute__((vector_size(32)));
#define AS1 __attribute__((address_space(1)))
#define AS3 __attribute__((address_space(3)))

#if defined(__AMDGCN__) && __has_builtin(__builtin_amdgcn_tensor_load_to_lds)
#define USE_TDM 1
#else
#define USE_TDM 0
#endif
#if defined(__AMDGCN__) && __has_builtin(__builtin_amdgcn_global_load_async_to_lds_b128)
#define USE_ASYNC_LDS 1
#else
#define USE_ASYNC_LDS 0
#endif

__global__ __launch_bounds__(NTHREADS)
void Conv2D_3x3_kernel(const float* __restrict__ X,
                       const float* __restrict__ Wt,
                       const float* __restrict__ Bs,
                       float* __restrict__ out)
{
    __shared__ float lds[IN_ROWS * LSTR];

    const int tid = threadIdx.x;
    const int gx  = blockIdx.x * TILE;   // input/output col base of tile
    const int gy  = blockIdx.y * TILE;   // input/output row base of tile

    // ---------------- Stage input tile (with halo) into LDS ----------------
#if USE_TDM
    // One TDM descriptor per tile: 2D tile of 68 x 66 fp32, contiguous rows in
    // LDS (tile_dim0 == LSTR). tensor_dim0/1 = remaining extent from the tile
    // origin (D#.global_addr is the *tile* start), so out-of-tensor reads on
    // edge tiles are zero-filled by the TDM engine.
    if (tid < 32) {                       // one wave issues (TDM ignores EXEC)
        const unsigned lbase = (unsigned)(unsigned long long)(AS3 void*)(void*)lds;
        const unsigned long long gaddr =
            (unsigned long long)(size_t)X +
            ((unsigned long long)gy * IN_W + (unsigned)gx) * 4ull;
        const unsigned tdim0 = (unsigned)(IN_W - gx);   // remaining cols
        const unsigned tdim1 = (unsigned)(IN_H - gy);   // remaining rows

        // D# group 0: [1:0]=count=1 | [63:32]=lds_addr | [120:64]=global_addr | [127:126]=type=2
        v4u g0 = { 1u,
                   lbase,
                   (unsigned)gaddr,
                   (unsigned)((gaddr >> 32) & 0x01ffffffull) | 0x80000000u };

        // D# group 1: [17:16]=data_size=2 (4B) | [79:48]=tensor_dim0 |
        // [111:80]=tensor_dim1 | [127:112]=tile_dim0 | [143:128]=tile_dim1 |
        // [207:160]=tensor_dim0_stride
        v8i g1 = { (int)0x00020000u,
                   (int)((tdim0 & 0xffffu) << 16),
                   (int)(((tdim0 >> 16) & 0xffffu) | ((tdim1 & 0xffffu) << 16)),
                   (int)(((tdim1 >> 16) & 0xffffu) | ((unsigned)LSTR << 16)),
                   (int)IN_ROWS,
                   (int)IN_W,
                   0, 0 };

        v4i g2 = { 0, 0, 0, 0 };          // dims 3/4 unused (tile_dim3/4 = 0)
        v4i g3 = { 0, 0, 0, 0 };
        v8i g4 = { 0, 0, 0, 0, 0, 0, 0, 0 };

        __builtin_amdgcn_tensor_load_to_lds(g0, g1, g2, g3, g4, /*cpol=*/0);
 #if __has_builtin(__builtin_amdgcn_s_wait_tensorcnt)
        __builtin_amdgcn_s_wait_tensorcnt(0);
 #else
        asm volatile("s_wait_tensorcnt 0x0" ::: "memory");
 #endif
    }
    asm volatile("" ::: "memory");        // LDS written by TDM, invisible to AA
#elif USE_ASYNC_LDS
    for (int id = tid; id < IN_ROWS * CHUNKS; id += NTHREADS) {
        const int row = id / CHUNKS;
        const int ch  = id - row * CHUNKS;
        const int gr  = gy + row;
        const int gc  = gx + (ch << 2);
        if (gr < IN_H && gc < IN_W) {     // chunks never partially OOB (4 | 4096)
            const float* gsrc = X + (size_t)gr * IN_W + gc;
            float*       ldst = &lds[row * LSTR + (ch << 2)];
            __builtin_amdgcn_global_load_async_to_lds_b128(
                (AS1 v4i*)(AS1 void*)(void*)gsrc,
                (AS3 v4i*)(AS3 void*)(void*)ldst,
                /*offset=*/0, /*cpol=*/0);
        }
    }
 #if __has_builtin(__builtin_amdgcn_s_wait_asynccnt)
    __builtin_amdgcn_s_wait_asynccnt(0);
 #else
    asm volatile("s_wait_asynccnt 0" ::: "memory");
 #endif
#else
    for (int id = tid; id < IN_ROWS * CHUNKS; id += NTHREADS) {
        const int row = id / CHUNKS;
        const int ch  = id - row * CHUNKS;
        const int gr  = gy + row;
        const int gc  = gx + (ch << 2);
        if (gr < IN_H && gc < IN_W) {
            *(float4*)&lds[row * LSTR + (ch << 2)] =
                *(const float4*)(X + (size_t)gr * IN_W + gc);
        }
    }
#endif
    __syncthreads();

    // ---------------- Uniform weights / bias (scalar loads) ----------------
    const float w00 = Wt[0], w01 = Wt[1], w02 = Wt[2];
    const float w10 = Wt[3], w11 = Wt[4], w12 = Wt[5];
    const float w20 = Wt[6], w21 = Wt[7], w22 = Wt[8];
    const float bias = Bs[0];

    // ---------------- Compute: 1 column x 16 rows per thread ----------------
    const int c     = tid & (TILE - 1);   // column within tile (0..63)
    const int rg    = tid >> 6;           // row group (0..3), 16 rows each
    const int ocol  = gx + c;
    const int orow0 = gy + rg * 16;
    const bool colok = (ocol < OUT_W);

    const float* ls = &lds[(rg * 16) * LSTR + c];

    // rolling 3-row x 3-col register window
    float a0 = ls[0],        a1 = ls[1],        a2 = ls[2];
    float b0 = ls[LSTR + 0], b1 = ls[LSTR + 1], b2 = ls[LSTR + 2];

    #pragma unroll
    for (int rr = 0; rr < 16; ++rr) {
        const float* l2 = ls + (rr + 2) * LSTR;
        const float c0 = l2[0], c1 = l2[1], c2 = l2[2];

        float acc = bias;
        acc = fmaf(w00, a0, acc);
        acc = fmaf(w01, a1, acc);
        acc = fmaf(w02, a2, acc);
        acc = fmaf(w10, b0, acc);
        acc = fmaf(w11, b1, acc);
        acc = fmaf(w12, b2, acc);
        acc = fmaf(w20, c0, acc);
        acc = fmaf(w21, c1, acc);
        acc = fmaf(w22, c2, acc);

        const int orow = orow0 + rr;
        if (colok && orow < OUT_H)
            out[(size_t)orow * OUT_W + ocol] = acc;

        a0 = b0; a1 = b1; a2 = b2;
        b0 = c0; b1 = c1; b2 = c2;
    }
}

extern "C" void kernel_launch(void* const* d_in, const int* in_sizes, int n_in,
                              void* d_out, int out_size, void* d_ws, size_t ws_size,
                              hipStream_t stream) {
    (void)in_sizes; (void)n_in; (void)out_size; (void)d_ws; (void)ws_size;
    const float* X  = (const float*)d_in[0];
    const float* Wt = (const float*)d_in[1];
    const float* Bs = (const float*)d_in[2];
    float* out      = (float*)d_out;

    dim3 grid((OUT_W + TILE - 1) / TILE, (OUT_H + TILE - 1) / TILE); // 64 x 64
    Conv2D_3x3_kernel<<<grid, NTHREADS, 0, stream>>>(X, Wt, Bs, out);
}